// SelfAttention_8040178778230
// MI455X (gfx1250) — compile-verified
//
#include <hip/hip_runtime.h>

// ---------------------------------------------------------------------------
// MI455X (gfx1250) decode-attention, fp32 end-to-end.
// Memory-bound problem (~675MB streamed, ~29us floor @ 23.3TB/s): all GEMMs
// run on V_WMMA_F32_16X16X4_F32, every HBM byte is streamed exactly once,
// and streaming loops use b128 loads via a K-slot permutation (WMMA's K
// reduction is order-independent, so hardware K-slots can carry permuted
// logical k as long as A and B agree).
// ---------------------------------------------------------------------------

typedef float v2f __attribute__((ext_vector_type(2)));
typedef float v4f __attribute__((ext_vector_type(4)));
typedef float v8f __attribute__((ext_vector_type(8)));

__device__ __forceinline__ v8f wmma_f32_4(v2f a, v2f b, v8f c) {
    // 8-arg form: (neg_a, A, neg_b, B, c_mod, C, reuse_a, reuse_b)
    return __builtin_amdgcn_wmma_f32_16x16x4_f32(false, a, false, b, (short)0, c,
                                                 false, false);
}

constexpr int B_    = 32;
constexpr int DIM   = 4096;
constexpr int HQ    = 32;
constexpr int HKV   = 8;
constexpr int HD    = 128;
constexpr int NREP  = 4;        // HQ / HKV
constexpr int SEQ   = 2048;     // start_pos + 1  (== MAX_S)
constexpr int QKV_N = HQ*HD + 2*HKV*HD;   // 6144 combined q|k|v columns

// ---------------------------------------------------------------------------
// Y[0:32, n0:n0+16] = X(32xK) @ W^T tile, via f32 WMMA 16x16x4.
// One wave per 16-wide N tile; both 16-row M tiles share one B load so each
// weight element is streamed from HBM exactly once. K=8 chunk per iteration:
// one b128 per operand feeds two WMMAs (K-slot permutation, see header).
// ---------------------------------------------------------------------------
__global__ void __launch_bounds__(256)
gemm32_xwT(const float* __restrict__ X, const float* __restrict__ W,
           float* __restrict__ Y, int N, int K, int ldY)
{
    const int wv = blockIdx.x * (blockDim.x >> 5) + (threadIdx.x >> 5);
    if (wv >= (N >> 4)) return;                 // wave-uniform exit
    const int n0   = wv << 4;
    const int lane = threadIdx.x & 31;
    const int l15  = lane & 15;
    const int half = lane >> 4;

    // per-lane rows; +4*half selects this half-wave's logical-k quartet
    const float* a0p = X + (size_t)l15        * K + 4*half;   // rows 0..15
    const float* a1p = X + (size_t)(l15 + 16) * K + 4*half;   // rows 16..31
    const float* bp  = W + (size_t)(n0 + l15) * K + 4*half;   // B[k][n]=W[n][k]

    v8f c0 = {};
    v8f c1 = {};
#pragma unroll 4
    for (int k = 0; k < K; k += 8) {
        v4f bv = *(const v4f*)(bp  + k);
        v4f a0 = *(const v4f*)(a0p + k);
        v4f a1 = *(const v4f*)(a1p + k);
        // WMMA_lo covers logical k {0,1,4,5}; WMMA_hi covers {2,3,6,7}
        c0 = wmma_f32_4(a0.xy, bv.xy, c0);
        c1 = wmma_f32_4(a1.xy, bv.xy, c1);
        c0 = wmma_f32_4(a0.zw, bv.zw, c0);
        c1 = wmma_f32_4(a1.zw, bv.zw, c1);
    }
    // C/D layout: VGPR r -> (M = r + 8*half, N = l15)
#pragma unroll
    for (int r = 0; r < 8; ++r) {
        const int m = r + half*8;
        Y[(size_t)m        * ldY + n0 + l15] = c0[r];
        Y[(size_t)(m + 16) * ldY + n0 + l15] = c1[r];
    }
}

// ---------------------------------------------------------------------------
// In-place RoPE on the q (cols 0..4095) and k (cols 4096..5119) sections of
// the combined 32x6144 QKV buffer. One thread per (real,imag) pair.
// ---------------------------------------------------------------------------
__global__ void rope_kernel(float* __restrict__ Y, const float* __restrict__ ang)
{
    const int NQ = B_ * (HQ*HD/2);      // 65536 q pairs
    const int NK = B_ * (HKV*HD/2);     // 16384 k pairs
    int i = blockIdx.x * blockDim.x + threadIdx.x;
    int b, col;
    if (i < NQ) {
        b = i >> 11; col = 2 * (i & 2047);
    } else {
        int j = i - NQ;
        if (j >= NK) return;
        b = j >> 9;  col = HQ*HD + 2 * (j & 511);
    }
    const int ai = (col >> 1) & 63;     // pair index within head (HD/2 = 64)
    const float a = ang[ai];
    const float c = cosf(a), s = sinf(a);
    float* yp = Y + (size_t)b * QKV_N + col;
    const float tr = yp[0], ti = yp[1];
    yp[0] = tr*c - ti*s;
    yp[1] = tr*s + ti*c;
}

// ---------------------------------------------------------------------------
// Flash-decode GQA attention. One block per (batch b, kv-head g); 8 waves,
// each streams 256 key positions with online softmax; both score and P@V
// GEMMs run on V_WMMA_F32_16X16X4_F32. Row t==2047 is served from the
// workspace (new k/v) via per-lane pointer selects.
// ---------------------------------------------------------------------------
__global__ void __launch_bounds__(256)
attn_decode(const float* __restrict__ ws_y, const float* __restrict__ cache_k,
            const float* __restrict__ cache_v, float* __restrict__ aout)
{
    const int b    = blockIdx.x >> 3;
    const int g    = blockIdx.x & 7;
    const int tid  = threadIdx.x;
    const int w    = tid >> 5;
    const int lane = tid & 31;
    const int l15  = lane & 15;
    const int half = lane >> 4;

    __shared__ __align__(16) float qs[NREP * HD];   // pre-scaled q (4 x 128)
    __shared__ float Olds[8 * NREP * HD];           // per-wave partial outputs
    __shared__ float mlds[8 * NREP];
    __shared__ float llds[8 * NREP];

    const float scale = 0.08838834764831845f;  // 1/sqrt(128)
    const float* qsrc = ws_y + (size_t)b * QKV_N + g * (NREP*HD);
    for (int i = tid; i < NREP*HD; i += blockDim.x) qs[i] = qsrc[i] * scale;
    __syncthreads();

    const float* kws = ws_y + (size_t)b * QKV_N + HQ*HD + g * HD;            // new k row
    const float* vws = ws_y + (size_t)b * QKV_N + HQ*HD + HKV*HD + g * HD;   // new v row

    float m_ln = -1e30f, l_ln = 0.f;
    v8f o[8];
#pragma unroll
    for (int dt = 0; dt < 8; ++dt) o[dt] = v8f{};

    const int tstart = w * 256;
    for (int t0 = tstart; t0 < tstart + 256; t0 += 16) {
        // ---- scores^T tile: D[t][h] = K(16t x kd) x q^T(kd x 16h) ----
        const int tk = t0 + l15;                                // A-row per lane
        const float* krow = ((tk == SEQ-1) ? kws
            : cache_k + (((size_t)b * SEQ + tk) * HKV + g) * HD) + 4*half;
        const float* qrow = &qs[(l15 & 3) * HD] + 4*half;       // heads padded 4->16
        v8f sc = {};
#pragma unroll
        for (int kd = 0; kd < HD; kd += 8) {
            v4f a  = *(const v4f*)(krow + kd);
            v4f bq = *(const v4f*)(qrow + kd);
            sc = wmma_f32_4(a.xy, bq.xy, sc);   // logical kd {0,1,4,5}
            sc = wmma_f32_4(a.zw, bq.zw, sc);   // logical kd {2,3,6,7}
        }

        // ---- online softmax; lane tracks head h = l15, t = r + 8*half ----
        float tmax = sc[0];
#pragma unroll
        for (int r = 1; r < 8; ++r) tmax = fmaxf(tmax, sc[r]);
        tmax = fmaxf(tmax, __shfl_xor(tmax, 16));
        const float mnew  = fmaxf(m_ln, tmax);
        const float alpha = __expf(m_ln - mnew);
        float p[8], rs = 0.f;
#pragma unroll
        for (int r = 0; r < 8; ++r) { p[r] = __expf(sc[r] - mnew); rs += p[r]; }
        rs += __shfl_xor(rs, 16);
        l_ln = l_ln * alpha + rs;
        m_ln = mnew;

        // rescale accumulator rows (row r == head r; rows 4..7 are padding)
        const float a0 = __shfl(alpha, 0), a1 = __shfl(alpha, 1);
        const float a2 = __shfl(alpha, 2), a3 = __shfl(alpha, 3);
#pragma unroll
        for (int dt = 0; dt < 8; ++dt) {
            o[dt][0] *= a0; o[dt][1] *= a1; o[dt][2] *= a2; o[dt][3] *= a3;
        }

        // ---- P @ V : D[h][d] += P(16h x 4t) x V(4t x 16d) ----
#pragma unroll
        for (int j = 0; j < 4; ++j) {
            const int base = (j & 1) * 4;
            const int hi   = j >> 1;
            // restage P (score layout: t = r + 8*half) into A layout
            // (K = 2*half + {0,1}) using one half-wave exchange
            const float s0 = __shfl_xor(p[base + (hi ? 0 : 2)], 16);
            const float s1 = __shfl_xor(p[base + (hi ? 1 : 3)], 16);
            float ax, ay;
            if (hi == 0) { ax = half ? s0 : p[base];      ay = half ? s1 : p[base+1]; }
            else         { ax = half ? p[base+2] : s0;    ay = half ? p[base+3] : s1; }
            v2f av; av.x = ax; av.y = ay;

            const int tb = t0 + 4*j + 2*half;               // B-rows tb, tb+1
            const float* vr0 = (tb == SEQ-1) ? vws
                : cache_v + (((size_t)b * SEQ + tb) * HKV + g) * HD;
            const float* vr1 = (tb + 1 == SEQ-1) ? vws
                : cache_v + (((size_t)b * SEQ + tb + 1) * HKV + g) * HD;
#pragma unroll
            for (int dt = 0; dt < 8; ++dt) {
                v2f bv; bv.x = vr0[dt*16 + l15]; bv.y = vr1[dt*16 + l15];
                o[dt] = wmma_f32_4(av, bv, o[dt]);
            }
        }
    }

    // ---- per-wave partials -> LDS (heads 0..3 live in lanes 0..15, r=0..3)
    if (half == 0) {
#pragma unroll
        for (int dt = 0; dt < 8; ++dt)
#pragma unroll
            for (int r = 0; r < 4; ++r)
                Olds[(w*4 + r) * HD + dt*16 + l15] = o[dt][r];
        if (l15 < 4) { mlds[w*4 + l15] = m_ln; llds[w*4 + l15] = l_ln; }
    }
    __syncthreads();

    // ---- combine 8 wave partials ----
    for (int e = tid; e < NREP*HD; e += blockDim.x) {
        const int h = e >> 7, d = e & 127;
        float M = mlds[h];
        for (int ww = 1; ww < 8; ++ww) M = fmaxf(M, mlds[ww*4 + h]);
        float L = 0.f, O = 0.f;
        for (int ww = 0; ww < 8; ++ww) {
            const float f = __expf(mlds[ww*4 + h] - M);
            L += llds[ww*4 + h] * f;
            O += Olds[(ww*4 + h) * HD + d] * f;
        }
        aout[(size_t)b * DIM + g * (NREP*HD) + e] = O / L;
    }
}

// ---------------------------------------------------------------------------
extern "C" void kernel_launch(void* const* d_in, const int* in_sizes, int n_in,
                              void* d_out, int out_size, void* d_ws, size_t ws_size,
                              hipStream_t stream) {
    const float* x   = (const float*)d_in[0];
    const float* ang = (const float*)d_in[1];
    const float* wq  = (const float*)d_in[2];
    const float* wk  = (const float*)d_in[3];
    const float* wv  = (const float*)d_in[4];
    const float* wo  = (const float*)d_in[5];
    const float* ck  = (const float*)d_in[6];
    const float* cv  = (const float*)d_in[7];
    float* out = (float*)d_out;

    float* y    = (float*)d_ws;                 // 32 x 6144 combined q|k|v
    float* aout = y + (size_t)B_ * QKV_N;       // 32 x 4096 attention output

    const dim3 blk(256);
    // QKV projections (one combined buffer, ld = 6144)
    gemm32_xwT<<<(HQ*HD)/16/8,  blk, 0, stream>>>(x, wq, y,                 HQ*HD,  DIM, QKV_N);
    gemm32_xwT<<<(HKV*HD)/16/8, blk, 0, stream>>>(x, wk, y + HQ*HD,         HKV*HD, DIM, QKV_N);
    gemm32_xwT<<<(HKV*HD)/16/8, blk, 0, stream>>>(x, wv, y + HQ*HD + HKV*HD,HKV*HD, DIM, QKV_N);
    // RoPE on q and k
    const int nrope = B_*(HQ*HD/2) + B_*(HKV*HD/2);
    rope_kernel<<<(nrope + 255)/256, blk, 0, stream>>>(y, ang);
    // flash-decode attention: one block per (b, kv-head)
    attn_decode<<<B_*HKV, blk, 0, stream>>>(y, ck, cv, aout);
    // output projection
    gemm32_xwT<<<DIM/16/8, blk, 0, stream>>>(aout, wo, out, DIM, DIM, DIM);
}